// ChamferLoss_20203526161089
// MI455X (gfx1250) — compile-verified
//
#include <hip/hip_runtime.h>

typedef __attribute__((ext_vector_type(2))) float v2f;
typedef __attribute__((ext_vector_type(8))) float v8f;

#define B_SZ 32
#define N_SZ 2048
#define M_TILES (N_SZ / 16)   // 128 m-tiles per batch
#define RT_PER_BLK 8          // 8 waves per block, one row-tile each

// ---------------------------------------------------------------------------
// Kernel 0: init per-(b,m) and per-(b,n) running minima to +inf.
// minN (min over n -> [B,M]) kept as u32 bit patterns for global_atomic_min_u32
// (all distances are >= 0, so float ordering == u32 bit-pattern ordering).
// minM (min over m -> [B,N]) plain floats (wave-exclusive stores).
// ---------------------------------------------------------------------------
__global__ void chamfer_init(unsigned int* __restrict__ minN,
                             float* __restrict__ minM) {
    int i = blockIdx.x * blockDim.x + threadIdx.x;
    if (i < B_SZ * N_SZ) {
        minN[i] = 0x7F800000u;              // +inf
        minM[i] = __builtin_huge_valf();    // +inf
    }
}

// ---------------------------------------------------------------------------
// Kernel 1: per-batch tiled distance matrix via V_WMMA_F32_16X16X4_F32.
// grid.x = B_SZ * (M_TILES / RT_PER_BLK) = 32 * 16 blocks, 256 threads each.
// Each wave owns 16 rows (n) of one batch and sweeps 128 column tiles (m),
// unrolled 2 tiles/iteration: 2 WMMAs + 1 fully-converged u32 atomic-min
// (lanes 0-15 cover tile0's 16 columns, lanes 16-31 cover tile1's).
// ---------------------------------------------------------------------------
__global__ void __launch_bounds__(256)
chamfer_tiles(const float* __restrict__ x, const float* __restrict__ y,
              unsigned int* __restrict__ minN, float* __restrict__ minM) {
    __shared__ float ys[N_SZ * 3];   // y[b] staged: 24 KB
    __shared__ float y2s[N_SZ];      // per-m squared norms: 8 KB

    const int b       = blockIdx.x / (M_TILES / RT_PER_BLK);
    const int rt_base = (blockIdx.x % (M_TILES / RT_PER_BLK)) * RT_PER_BLK;

    const float* xb = x + b * (N_SZ * 3);
    const float* yb = y + b * (N_SZ * 3);

    // Cooperative stage of y[b] into LDS, then y^2.
    for (int i = threadIdx.x; i < N_SZ * 3; i += 256) ys[i] = yb[i];
    __syncthreads();
    for (int i = threadIdx.x; i < N_SZ; i += 256) {
        float a0 = ys[i * 3 + 0], a1 = ys[i * 3 + 1], a2 = ys[i * 3 + 2];
        y2s[i] = a0 * a0 + a1 * a1 + a2 * a2;
    }
    __syncthreads();

    const int wave = threadIdx.x >> 5;           // 0..7
    const int lane = threadIdx.x & 31;
    const int half = lane >> 4;                  // 0: K=0/1, 1: K=2/3
    const int col  = lane & 15;                  // tile-local column / row id
    const int n0   = (rt_base + wave) * 16;      // first n of this wave's tile

    // A-matrix (16x4 f32, ISA 7.12.2): lane<16 -> {K0,K1}, lane>=16 -> {K2,K3=0}.
    const int rowA = n0 + col;
    float a_k_lo = xb[rowA * 3 + (half ? 2 : 0)];
    float a_k_hi = xb[rowA * 3 + 1];
    v2f amat;
    amat[0] = a_k_lo;
    amat[1] = half ? 0.0f : a_k_hi;

    // x^2 for the 8 rows this lane's C VGPRs map to (M = v + 8*half).
    float x2v[8];
#pragma unroll
    for (int v = 0; v < 8; ++v) {
        int r = n0 + v + 8 * half;
        float r0 = xb[r * 3 + 0], r1 = xb[r * 3 + 1], r2 = xb[r * 3 + 2];
        x2v[v] = r0 * r0 + r1 * r1 + r2 * r2;
    }

    float rowmin[8];
#pragma unroll
    for (int v = 0; v < 8; ++v) rowmin[v] = __builtin_huge_valf();

    unsigned int* minNb = minN + b * N_SZ;
    const int ksel = half ? 2 : 0;

    for (int mt = 0; mt < M_TILES; mt += 2) {
        const int m0 = mt * 16 + col;        // column in tile 0
        const int m1 = m0 + 16;              // column in tile 1

        // B-matrix (4x16 f32): lane<16 -> {K0,K1} of column m, lane>=16 -> {K2,0}.
        float b0lo = ys[m0 * 3 + ksel];
        float b0hi = ys[m0 * 3 + 1];
        float b1lo = ys[m1 * 3 + ksel];
        float b1hi = ys[m1 * 3 + 1];
        v2f bmat0, bmat1;
        bmat0[0] = b0lo; bmat0[1] = half ? 0.0f : b0hi;
        bmat1[0] = b1lo; bmat1[1] = half ? 0.0f : b1hi;
        const float y2_0 = y2s[m0];
        const float y2_1 = y2s[m1];

        v8f c0 = {}, c1 = {};
        c0 = __builtin_amdgcn_wmma_f32_16x16x4_f32(
            false, amat, false, bmat0, (short)0, c0, false, false);
        c1 = __builtin_amdgcn_wmma_f32_16x16x4_f32(
            false, amat, false, bmat1, (short)0, c1, false, false);

        float colmin0 = __builtin_huge_valf();
        float colmin1 = __builtin_huge_valf();
#pragma unroll
        for (int v = 0; v < 8; ++v) {
            float d0 = fmaxf(x2v[v] + y2_0 - 2.0f * c0[v], 0.0f);
            float d1 = fmaxf(x2v[v] + y2_1 - 2.0f * c1[v], 0.0f);
            rowmin[v] = fminf(rowmin[v], fminf(d0, d1));
            colmin0 = fminf(colmin0, d0);
            colmin1 = fminf(colmin1, d1);
        }
        // Combine the two lane halves (rows 0..7 with rows 8..15); afterwards
        // every lane holds the full 16-row column min of both tiles.
        colmin0 = fminf(colmin0, __shfl_xor(colmin0, 16, 32));
        colmin1 = fminf(colmin1, __shfl_xor(colmin1, 16, 32));

        // Fully converged atomic: lane half selects which tile's column it
        // commits, so 32 lanes hit 32 distinct addresses — no exec masking.
        const int   m_sel = half ? m1 : m0;
        const float c_sel = half ? colmin1 : colmin0;
        atomicMin(&minNb[m_sel], __float_as_uint(c_sel));
    }

    // Row-min tree across the 16 lanes of each half.
#pragma unroll
    for (int v = 0; v < 8; ++v) {
        float r = rowmin[v];
        r = fminf(r, __shfl_xor(r, 1, 32));
        r = fminf(r, __shfl_xor(r, 2, 32));
        r = fminf(r, __shfl_xor(r, 4, 32));
        r = fminf(r, __shfl_xor(r, 8, 32));
        rowmin[v] = r;
    }
    if (col == 0) {
        float* minMb = minM + b * N_SZ;
#pragma unroll
        for (int v = 0; v < 8; ++v)
            minMb[n0 + v + 8 * half] = rowmin[v];   // wave-exclusive rows
    }
}

// ---------------------------------------------------------------------------
// Kernel 2: deterministic single-block tree reduction -> loss scalar.
// ---------------------------------------------------------------------------
__global__ void __launch_bounds__(1024)
chamfer_reduce(const unsigned int* __restrict__ minN,
               const float* __restrict__ minM, float* __restrict__ out) {
    __shared__ float sm[1024];
    float s = 0.0f;
    for (int i = threadIdx.x; i < B_SZ * N_SZ; i += 1024)
        s += __uint_as_float(minN[i]) + minM[i];
    sm[threadIdx.x] = s;
    __syncthreads();
    for (int st = 512; st > 0; st >>= 1) {
        if (threadIdx.x < st) sm[threadIdx.x] += sm[threadIdx.x + st];
        __syncthreads();
    }
    if (threadIdx.x == 0) out[0] = sm[0] / (float)B_SZ;
}

extern "C" void kernel_launch(void* const* d_in, const int* in_sizes, int n_in,
                              void* d_out, int out_size, void* d_ws, size_t ws_size,
                              hipStream_t stream) {
    const float* x = (const float*)d_in[0];
    const float* y = (const float*)d_in[1];
    float* out = (float*)d_out;

    unsigned int* minN = (unsigned int*)d_ws;                 // B*N u32
    float* minM = (float*)((char*)d_ws + B_SZ * N_SZ * 4);    // B*N f32

    chamfer_init<<<(B_SZ * N_SZ + 255) / 256, 256, 0, stream>>>(minN, minM);
    chamfer_tiles<<<B_SZ * (M_TILES / RT_PER_BLK), 256, 0, stream>>>(x, y, minN, minM);
    chamfer_reduce<<<1, 1024, 0, stream>>>(minN, minM, out);
}